// TriangleAttentionStartingNode_52183852646477
// MI455X (gfx1250) — compile-verified
//
#include <hip/hip_runtime.h>

// TriangleAttentionStartingNode: B=1, N=256x256, C=32, H=4, all fp32.
// Strategy: fp32 WMMA (v_wmma_f32_16x16x4_f32) everywhere to keep reference
// precision; problem is HBM-bound (~50MB @ 23.3TB/s), so f32 matrix ops are
// free. Kernel1: one WG per (row i, head h), LDS-resident zn/k/v. Kernel2:
// final 128->32 projection.

typedef float v2f __attribute__((ext_vector_type(2)));
typedef float v8f __attribute__((ext_vector_type(8)));

#define ST 36  // LDS row stride (floats); 36 mod 64 cycles 16 distinct banks

static __device__ inline v8f wmma4(v2f a, v2f b, v8f c) {
  // (neg_a, A, neg_b, B, c_mod, C, reuse_a, reuse_b)
  return __builtin_amdgcn_wmma_f32_16x16x4_f32(false, a, false, b, (short)0, c,
                                               false, false);
}

__global__ __launch_bounds__(256) void tri_attn_row(
    const float* __restrict__ z, const float* __restrict__ ln_g,
    const float* __restrict__ ln_b, const float* __restrict__ Wq,
    const float* __restrict__ Wk, const float* __restrict__ Wv,
    const float* __restrict__ Wg, const float* __restrict__ bg,
    float* __restrict__ ows) {
  extern __shared__ float smem[];
  float* zn = smem;                 // 256 x ST
  float* kbuf = zn + 256 * ST;      // 256 x ST (this head's K, 32 cols used)
  float* vbuf = kbuf + 256 * ST;    // 256 x ST (this head's V)
  float* scr = vbuf + 256 * ST;     // 8 waves x 16 x ST transpose scratch

  const int tid = threadIdx.x;
  const int lane = tid & 31;
  const int wid = tid >> 5;   // wave id 0..7
  const int l15 = lane & 15;  // row/col within tile
  const int hi = lane >> 4;   // K-pair select (0 or 1)
  const int bi = blockIdx.x;  // i-row
  const int h = blockIdx.y;   // head

  // ---- Phase 1: LayerNorm, one thread per position j = tid ----
  {
    const float* zr = z + ((size_t)bi * 256 + tid) * 32;
    float x[32];
    float s = 0.f;
#pragma unroll
    for (int c = 0; c < 32; ++c) {
      x[c] = zr[c];
      s += x[c];
    }
    const float mu = s * (1.f / 32.f);
    float vs = 0.f;
#pragma unroll
    for (int c = 0; c < 32; ++c) {
      float d = x[c] - mu;
      vs += d * d;
    }
    const float rs = rsqrtf(vs * (1.f / 32.f) + 1e-5f);
#pragma unroll
    for (int c = 0; c < 32; ++c)
      zn[tid * ST + c] = (x[c] - mu) * rs * ln_g[c] + ln_b[c];
  }
  __syncthreads();

  // ---- Phase 2: K,V projections for this head into LDS ----
  // 64 jobs: {mat(2)} x {mtile(16)} x {ntile(2)}, 8 per wave.
  for (int job = wid; job < 64; job += 8) {
    const int mat = job & 1;
    const int mt = (job >> 1) & 15;
    const int nt = job >> 5;
    const float* W = mat ? Wv : Wk;
    const float* Wrow = W + (size_t)(h * 32 + nt * 16 + l15) * 32;
    float* dst = mat ? vbuf : kbuf;
    v8f acc = {};
#pragma unroll
    for (int kk = 0; kk < 32; kk += 4) {
      v2f a = *(const v2f*)&zn[(mt * 16 + l15) * ST + kk + 2 * hi];
      v2f b = *(const v2f*)&Wrow[kk + 2 * hi];
      acc = wmma4(a, b, acc);
    }
#pragma unroll
    for (int r = 0; r < 8; ++r)
      dst[(mt * 16 + r + 8 * hi) * ST + nt * 16 + l15] = acc[r];
  }
  __syncthreads();

  // ---- Phase 3: attention. Each wave owns 2 q-tiles. ----
  float* qs = scr + wid * 16 * ST;  // per-wave 16xST scratch
  const float qscale = 0.17677669529663687f;  // C^-0.5

  for (int p = 0; p < 2; ++p) {
    const int qt = wid * 2 + p;

    // Q tile (pre-scaled) into per-wave scratch
#pragma unroll
    for (int nt = 0; nt < 2; ++nt) {
      v8f acc = {};
      const float* Wrow = Wq + (size_t)(h * 32 + nt * 16 + l15) * 32;
#pragma unroll
      for (int kk = 0; kk < 32; kk += 4) {
        v2f a = *(const v2f*)&zn[(qt * 16 + l15) * ST + kk + 2 * hi];
        v2f b = *(const v2f*)&Wrow[kk + 2 * hi];
        acc = wmma4(a, b, acc);
      }
#pragma unroll
      for (int r = 0; r < 8; ++r)
        qs[(r + 8 * hi) * ST + nt * 16 + l15] = acc[r] * qscale;
    }

    // Scores S[qt-rows][all 256 keys]: 16 key-tiles, K=32.
    // Bias b (zn@Wb.T) is constant over the softmax axis -> dropped.
    v8f S[16];
#pragma unroll
    for (int t = 0; t < 16; ++t) {
      v8f acc = {};
#pragma unroll
      for (int kk = 0; kk < 32; kk += 4) {
        v2f a = *(const v2f*)&qs[l15 * ST + kk + 2 * hi];
        v2f b = *(const v2f*)&kbuf[(t * 16 + l15) * ST + kk + 2 * hi];
        acc = wmma4(a, b, acc);
      }
      S[t] = acc;
    }

    // Softmax over keys. Row M = qt*16 + r + 8*hi; the 256 key values for a
    // row live across 16 lanes (same half) x 16 tiles at fixed r.
    float rsum[8];
#pragma unroll
    for (int r = 0; r < 8; ++r) {
      float m = S[0][r];
#pragma unroll
      for (int t = 1; t < 16; ++t) m = fmaxf(m, S[t][r]);
#pragma unroll
      for (int msk = 1; msk < 16; msk <<= 1)
        m = fmaxf(m, __shfl_xor(m, msk, 32));
      float s = 0.f;
#pragma unroll
      for (int t = 0; t < 16; ++t) {
        float e = __expf(S[t][r] - m);
        S[t][r] = e;
        s += e;
      }
#pragma unroll
      for (int msk = 1; msk < 16; msk <<= 1) s += __shfl_xor(s, msk, 32);
      rsum[r] = s;
    }

    // P @ V: per key-tile, transpose P through scratch (same-wave LDS is
    // in-order, no barrier needed), then K=16 as 4 wmma steps per n-tile.
    v8f O0 = {}, O1 = {};
#pragma unroll
    for (int t = 0; t < 16; ++t) {
#pragma unroll
      for (int r = 0; r < 8; ++r) qs[(r + 8 * hi) * ST + l15] = S[t][r];
#pragma unroll
      for (int kk = 0; kk < 16; kk += 4) {
        v2f a = *(const v2f*)&qs[l15 * ST + kk + 2 * hi];
        const int krow = t * 16 + kk + 2 * hi;
        v2f b0, b1;
        b0.x = vbuf[krow * ST + l15];
        b0.y = vbuf[(krow + 1) * ST + l15];
        b1.x = vbuf[krow * ST + 16 + l15];
        b1.y = vbuf[(krow + 1) * ST + 16 + l15];
        O0 = wmma4(a, b0, O0);
        O1 = wmma4(a, b1, O1);
      }
    }

    // Normalize, gate (sigmoid(zn@Wg.T+bg) via wmma; D-layout matches O),
    // and store gated o to workspace: [pos][h*32+c], fp32.
#pragma unroll
    for (int nt = 0; nt < 2; ++nt) {
      v8f O = nt ? O1 : O0;
      v8f gacc = {};
      const float* Wrow = Wg + (size_t)(h * 32 + nt * 16 + l15) * 32;
      const float bgv = bg[h * 32 + nt * 16 + l15];
#pragma unroll
      for (int kk = 0; kk < 32; kk += 4) {
        v2f a = *(const v2f*)&zn[(qt * 16 + l15) * ST + kk + 2 * hi];
        v2f b = *(const v2f*)&Wrow[kk + 2 * hi];
        gacc = wmma4(a, b, gacc);
      }
#pragma unroll
      for (int r = 0; r < 8; ++r) {
        const float gate = 1.f / (1.f + __expf(-(gacc[r] + bgv)));
        const float val = (O[r] / rsum[r]) * gate;
        const int m = qt * 16 + r + 8 * hi;
        ows[((size_t)bi * 256 + m) * 128 + h * 32 + nt * 16 + l15] = val;
      }
    }
  }
}

// Final projection: out = o(65536x128) @ Wo.T(128x32) + bo
__global__ __launch_bounds__(256) void tri_attn_oproj(
    const float* __restrict__ o, const float* __restrict__ Wo,
    const float* __restrict__ bo, float* __restrict__ out) {
  const int tid = threadIdx.x;
  const int lane = tid & 31;
  const int wid = tid >> 5;
  const int l15 = lane & 15;
  const int hi = lane >> 4;
  const int mt = blockIdx.x * 8 + wid;  // 4096 M-tiles total

  const float* Arow = o + ((size_t)mt * 16 + l15) * 128;
#pragma unroll
  for (int nt = 0; nt < 2; ++nt) {
    v8f acc = {};
    const float* Brow = Wo + (size_t)(nt * 16 + l15) * 128;
    const float bov = bo[nt * 16 + l15];
#pragma unroll
    for (int kk = 0; kk < 128; kk += 4) {
      v2f a = *(const v2f*)&Arow[kk + 2 * hi];
      v2f b = *(const v2f*)&Brow[kk + 2 * hi];
      acc = wmma4(a, b, acc);
    }
#pragma unroll
    for (int r = 0; r < 8; ++r)
      out[((size_t)mt * 16 + r + 8 * hi) * 32 + nt * 16 + l15] = acc[r] + bov;
  }
}

extern "C" void kernel_launch(void* const* d_in, const int* in_sizes, int n_in,
                              void* d_out, int out_size, void* d_ws,
                              size_t ws_size, hipStream_t stream) {
  (void)in_sizes;
  (void)n_in;
  (void)out_size;
  (void)ws_size;
  const float* z = (const float*)d_in[0];
  const float* ln_g = (const float*)d_in[1];
  const float* ln_b = (const float*)d_in[2];
  const float* Wq = (const float*)d_in[3];
  const float* Wk = (const float*)d_in[4];
  const float* Wv = (const float*)d_in[5];
  // d_in[6] = Wb: bias is constant over softmax axis => no-op, unused.
  const float* Wg = (const float*)d_in[7];
  const float* bg = (const float*)d_in[8];
  const float* Wo = (const float*)d_in[9];
  const float* bo = (const float*)d_in[10];
  float* ows = (float*)d_ws;  // 65536 x 128 fp32 gated attention output
  float* out = (float*)d_out;

  const int smem_bytes = (3 * 256 * ST + 8 * 16 * ST) * (int)sizeof(float);
  hipFuncSetAttribute((const void*)tri_attn_row,
                      hipFuncAttributeMaxDynamicSharedMemorySize, smem_bytes);

  dim3 g1(256, 4);  // (i-row, head)
  tri_attn_row<<<g1, 256, smem_bytes, stream>>>(z, ln_g, ln_b, Wq, Wk, Wv, Wg,
                                                bg, ows);
  tri_attn_oproj<<<512, 256, 0, stream>>>(ows, Wo, bo, out);
}